// Biaffine_57870389346348
// MI455X (gfx1250) — compile-verified
//
#include <hip/hip_runtime.h>
#include <math.h>

typedef __attribute__((ext_vector_type(2))) float v2f;
typedef __attribute__((ext_vector_type(4))) float v4f;
typedef __attribute__((ext_vector_type(8))) float v8f;

namespace {
constexpr int kB   = 4;
constexpr int kL   = 256;
constexpr int kD   = 512;
constexpr int kO   = 128;
constexpr int kDP1 = 513;     // D+1 (augmented dim)
constexpr int kUS  = 514;     // even LDS stride for Ueff rows (8B-aligned b64 loads)
}

__launch_bounds__(128, 4)
__global__ void biaffine_fused_kernel(const float* __restrict__ x1,
                                      const float* __restrict__ x2,
                                      const float* __restrict__ bw,
                                      const float* __restrict__ Wb,   // [O, 513, 513]
                                      const float* __restrict__ Wl,   // [1024, O]
                                      const float* __restrict__ bl,   // [O]
                                      float* __restrict__ out) {      // [B, L, L, O]
    __shared__ float x1s[16 * kD];   // 32 KB: x1 tile, rows x0..x0+15
    __shared__ float Us[16 * kUS];   // ~32.9 KB: Ueff[16][513] (+pad)

    const int tid  = threadIdx.x;
    const int lane = tid & 31;
    const int wave = tid >> 5;       // 0..3
    const int hl   = lane >> 4;      // half-wave: selects K-pair
    const int ln   = lane & 15;      // M/N index within fragment

    // block -> (b, o, x-tile)
    int bi = blockIdx.x;
    const int xt = bi & 15;  bi >>= 4;          // L/16 = 16 x-tiles
    const int o  = bi & (kO - 1);
    const int b  = bi >> 7;
    const int x0 = xt * 16;

    // softmax over the 2 branch weights
    const float e0 = __expf(bw[0]);
    const float e1 = __expf(bw[1]);
    const float w0 = e0 / (e0 + e1);
    const float w1 = e1 / (e0 + e1);

    const float* __restrict__ Wo = Wb + (size_t)o * kDP1 * kDP1;

    // ---- cooperative coalesced load of the x1 tile (16 rows x 512) ----
    const float* __restrict__ x1base = x1 + ((size_t)(b * kL + x0)) * kD;
    #pragma unroll
    for (int k = 0; k < 16; ++k) {
        const int e = (k * 128 + tid) * 4;
        *(v4f*)&x1s[e] = *(const v4f*)&x1base[e];
    }
    // init bias column accumulator: w0*W[o,512,512] + w1*b_lin[o]
    if (tid < 16) {
        Us[tid * kUS + 512] = w0 * Wo[(size_t)512 * kDP1 + 512] + w1 * bl[o];
    }
    __syncthreads();

    // ---- bias column 512: Ueff[m,512] += sum_i x1[m,i]*(w0*W[o,i,512] + w1*Wl[i,o]) ----
    {
        const int m = tid & 15, part = tid >> 4;   // 8 partial ranges of 64
        float t = 0.f;
        const float* __restrict__ wcol = Wo + 512;
        for (int i = part * 64; i < part * 64 + 64; ++i) {
            t += x1s[m * kD + i] * (w0 * wcol[(size_t)i * kDP1] + w1 * Wl[(size_t)i * kO + o]);
        }
        atomicAdd(&Us[m * kUS + 512], t);          // ds_add_f32
    }

    // ---- Stage 1: Ueff[16 x 512] = w0*(x1 @ W_bil[o]) + w0*biasrow + w1*W_lin[512+j,o] ----
    for (int jt = wave; jt < 32; jt += 4) {        // 8 j-tiles per wave
        const int j0 = jt * 16;
        v8f acc = {};
        for (int ib = 0; ib < kD; ib += 32) {
            // prefetch W stream 32 rows ahead (speculative; dropped if unmapped)
            __builtin_prefetch(Wo + (size_t)(ib + 32 + 2 * hl) * kDP1 + j0 + ln, 0, 0);
            #pragma unroll
            for (int i0 = ib; i0 < ib + 32; i0 += 4) {
                v2f a, bf;
                a.x = x1s[ln * kD + i0 + 2 * hl];          // A[m=ln][k=2hl]   (ds b64)
                a.y = x1s[ln * kD + i0 + 2 * hl + 1];
                const float* wp = Wo + (size_t)(i0 + 2 * hl) * kDP1 + j0 + ln;
                bf.x = wp[0];                              // B[k=2hl][n=ln]   (coalesced)
                bf.y = wp[kDP1];
                acc = __builtin_amdgcn_wmma_f32_16x16x4_f32(
                        false, a, false, bf, (short)0, acc, false, false);
            }
        }
        const float bias = w0 * Wo[(size_t)512 * kDP1 + j0 + ln]
                         + w1 * Wl[(size_t)(kD + j0 + ln) * kO + o];
        #pragma unroll
        for (int r = 0; r < 8; ++r) {
            const int m = r + 8 * hl;                      // D layout: m=r+8*hl, n=ln
            Us[m * kUS + j0 + ln] = w0 * acc[r] + bias;
        }
    }
    __syncthreads();

    // ---- Stage 2: out[x-tile, y, o] = Ueff @ x2^T + Ueff[:,512] ----
    const float* __restrict__ x2base = x2 + (size_t)b * kL * kD;
    float* __restrict__ outbase = out + (((size_t)b * kL + x0) * kL) * kO + o;
    for (int yt = wave; yt < 16; yt += 4) {        // 4 y-tiles per wave
        const int y0 = yt * 16;
        v8f acc = {};
        for (int jb = 0; jb < kD; jb += 32) {
            #pragma unroll
            for (int j0 = jb; j0 < jb + 32; j0 += 4) {
                v2f a, bf;
                a.x = Us[ln * kUS + j0 + 2 * hl];          // A[m=ln][k] from LDS (b64)
                a.y = Us[ln * kUS + j0 + 2 * hl + 1];
                const float* xp = x2base + (size_t)(y0 + ln) * kD + j0 + 2 * hl;
                bf.x = xp[0];                              // B[k][n=ln] = x2[y0+ln, j0+k]
                bf.y = xp[1];
                acc = __builtin_amdgcn_wmma_f32_16x16x4_f32(
                        false, a, false, bf, (short)0, acc, false, false);
            }
        }
        #pragma unroll
        for (int r = 0; r < 8; ++r) {
            const int m = r + 8 * hl;
            const float v = acc[r] + Us[m * kUS + 512];    // add fused bias column
            outbase[((size_t)m * kL + y0 + ln) * kO] = v;  // RT stores -> L2 write-combine
        }
    }
}

extern "C" void kernel_launch(void* const* d_in, const int* in_sizes, int n_in,
                              void* d_out, int out_size, void* d_ws, size_t ws_size,
                              hipStream_t stream) {
    const float* x1 = (const float*)d_in[0];
    const float* x2 = (const float*)d_in[1];
    const float* bw = (const float*)d_in[2];
    const float* Wb = (const float*)d_in[3];
    const float* Wl = (const float*)d_in[4];
    const float* bl = (const float*)d_in[5];
    float* out = (float*)d_out;

    const int grid = kB * kO * (kL / 16);   // 8192 blocks
    biaffine_fused_kernel<<<grid, 128, 0, stream>>>(x1, x2, bw, Wb, Wl, bl, out);
}